// RoutingFunction_28235115003998
// MI455X (gfx1250) — compile-verified
//
#include <hip/hip_runtime.h>
#include <hip/hip_bf16.h>
#include <math.h>

#define BATCH    64
#define DIM      768
#define HW       3136     // 56*56
#define FREQ_DIM 256
#define NE       16       // num experts

typedef __attribute__((ext_vector_type(2))) float v2f;
typedef __attribute__((ext_vector_type(4))) float v4f;
typedef __attribute__((ext_vector_type(8))) float v8f;

// ---------------------------------------------------------------------------
// Kernel A: pooled[b*DIM+d] = mean over 56x56 spatial of x[b][d]
// One 256-thread block per row; b128 NT loads (stream-once, bypass cache
// retention); LDS tree reduce.  This stage is the entire roofline cost.
// ---------------------------------------------------------------------------
__global__ __launch_bounds__(256) void pool_kernel(const float* __restrict__ x,
                                                   float* __restrict__ pooled) {
    const int row = blockIdx.x;  // 0 .. BATCH*DIM-1
    const v4f* rp = reinterpret_cast<const v4f*>(x + (size_t)row * HW);
    float s = 0.f;
    #pragma unroll 4
    for (int i = threadIdx.x; i < HW / 4; i += 256) {
        v4f v = __builtin_nontemporal_load(&rp[i]);   // global_load_b128, TH=NT
        s += v.x + v.y + v.z + v.w;
    }
    __shared__ float red[256];
    red[threadIdx.x] = s;
    __syncthreads();
    for (int off = 128; off > 0; off >>= 1) {
        if (threadIdx.x < off) red[threadIdx.x] += red[threadIdx.x + off];
        __syncthreads();
    }
    if (threadIdx.x == 0) pooled[row] = red[0] * (1.0f / (float)HW);
}

// ---------------------------------------------------------------------------
// Kernel B: logits(64x16) = pooled(64x768) @ gate_w^T + freq(64x256) @ fgw^T
// 4 waves, each owns one 16x16 M-tile, V_WMMA_F32_16X16X4_F32 over the K dim.
// f32 A 16x4 layout: lanes 0-15 -> {K=k0, K=k0+1}, lanes 16-31 -> {k0+2, k0+3}.
// C/D layout: VGPR j -> M = j (lanes 0-15) / M = j+8 (lanes 16-31), N = lane&15.
// ---------------------------------------------------------------------------
__global__ __launch_bounds__(128) void logits_kernel(const float* __restrict__ pooled,
                                                     const float* __restrict__ freq_emb,
                                                     const float* __restrict__ gate_w,
                                                     const float* __restrict__ freq_gate_w,
                                                     float* __restrict__ logits) {
    const int wave = threadIdx.x >> 5;
    const int lane = threadIdx.x & 31;
    const int m0   = wave * 16;
    const int mrow = m0 + (lane & 15);
    const int half = lane >> 4;        // 0: K pair {0,1}, 1: K pair {2,3}
    const int n    = lane & 15;

    v8f c = {0.f, 0.f, 0.f, 0.f, 0.f, 0.f, 0.f, 0.f};

    // pooled @ gate_w^T   (K = 768)
    for (int k0 = 0; k0 < DIM; k0 += 4) {
        const int kb = k0 + half * 2;
        v2f a, b;
        a.x = pooled[mrow * DIM + kb];
        a.y = pooled[mrow * DIM + kb + 1];
        b.x = gate_w[n * DIM + kb];        // B[k][n] = gate_w[n][k]
        b.y = gate_w[n * DIM + kb + 1];
        c = __builtin_amdgcn_wmma_f32_16x16x4_f32(false, a, false, b,
                                                  (short)0, c, false, false);
    }
    // freq_emb @ freq_gate_w^T   (K = 256)
    for (int k0 = 0; k0 < FREQ_DIM; k0 += 4) {
        const int kb = k0 + half * 2;
        v2f a, b;
        a.x = freq_emb[mrow * FREQ_DIM + kb];
        a.y = freq_emb[mrow * FREQ_DIM + kb + 1];
        b.x = freq_gate_w[n * FREQ_DIM + kb];
        b.y = freq_gate_w[n * FREQ_DIM + kb + 1];
        c = __builtin_amdgcn_wmma_f32_16x16x4_f32(false, a, false, b,
                                                  (short)0, c, false, false);
    }
    #pragma unroll
    for (int j = 0; j < 8; ++j) {
        const int row = m0 + j + half * 8;
        logits[row * NE + n] = c[j];
    }
}

// ---------------------------------------------------------------------------
// Kernel C: softmax / top-k / gates / aux losses.  One block, thread b owns
// batch row b; LDS ds_add_f32 atomics for the two cross-batch 16-vectors.
// ---------------------------------------------------------------------------
__global__ __launch_bounds__(64) void routing_kernel(const float* __restrict__ logits,
                                                     const float* __restrict__ complexity,
                                                     const float* __restrict__ noise,
                                                     const int*   __restrict__ kptr,
                                                     float* __restrict__ out) {
    __shared__ float imp[NE];    // sum_b softmax_clean[b][e]
    __shared__ float psum[NE];   // sum_b p[b][e]
    const int b = threadIdx.x;
    if (b < NE) { imp[b] = 0.f; psum[b] = 0.f; }
    __syncthreads();

    const int   kk        = min(max(kptr[0], 1), NE);
    const float noise_std = 1.0f / (float)NE;

    float lg[NE], noisy[NE], gs[NE], ex[NE];
    #pragma unroll
    for (int e = 0; e < NE; ++e) lg[e] = logits[b * NE + e];

    // clean softmax -> importance accumulation
    float mx = -1e30f;
    #pragma unroll
    for (int e = 0; e < NE; ++e) mx = fmaxf(mx, lg[e]);
    float sum = 0.f;
    #pragma unroll
    for (int e = 0; e < NE; ++e) { ex[e] = __expf(lg[e] - mx); sum += ex[e]; }
    float inv = 1.0f / sum;
    #pragma unroll
    for (int e = 0; e < NE; ++e) atomicAdd(&imp[e], ex[e] * inv);

    // noisy softmax
    #pragma unroll
    for (int e = 0; e < NE; ++e) noisy[e] = lg[e] + noise[b * NE + e] * noise_std;
    mx = -1e30f;
    #pragma unroll
    for (int e = 0; e < NE; ++e) mx = fmaxf(mx, noisy[e]);
    sum = 0.f;
    #pragma unroll
    for (int e = 0; e < NE; ++e) { gs[e] = __expf(noisy[e] - mx); sum += gs[e]; }
    inv = 1.0f / sum;
    #pragma unroll
    for (int e = 0; e < NE; ++e) gs[e] *= inv;

    // selection top-k (strict > scanning from e=0 == JAX lowest-index tiebreak);
    // softmax is monotone, so top-k(gs) indices == top-k(noisy) indices and the
    // k-th noisy threshold is noisy[idx[kk-1]].
    bool used[NE];
    #pragma unroll
    for (int e = 0; e < NE; ++e) used[e] = false;
    int idx[NE]; float val[NE];
    for (int j = 0; j < kk; ++j) {
        float best = -1e30f; int bi = 0;
        #pragma unroll
        for (int e = 0; e < NE; ++e)
            if (!used[e] && gs[e] > best) { best = gs[e]; bi = e; }
        used[bi] = true; idx[j] = bi; val[j] = best;
    }
    const float thr = noisy[idx[kk - 1]];

    // load probability p = 1 - Phi((thr - logit)/noise_std) = 0.5*erfc(z/sqrt2)
    #pragma unroll
    for (int e = 0; e < NE; ++e) {
        float z = (thr - lg[e]) * (float)NE;
        float p = 0.5f * erfcf(z * 0.70710678118654752f);
        atomicAdd(&psum[e], p);
    }

    // outputs: gates | indices(float) | values | aux_loss
    #pragma unroll
    for (int e = 0; e < NE; ++e) out[b * NE + e] = 0.f;
    for (int j = 0; j < kk; ++j) out[b * NE + idx[j]] = val[j];
    float* oidx = out + BATCH * NE;
    float* oval = oidx + BATCH * kk;
    for (int j = 0; j < kk; ++j) {
        oidx[b * kk + j] = (float)idx[j];
        oval[b * kk + j] = val[j];
    }

    __syncthreads();
    if (b == 0) {
        float v0[NE], v1[NE];
        float m0 = 0.f, m1 = 0.f;
        #pragma unroll
        for (int e = 0; e < NE; ++e) {
            v0[e] = imp[e] * complexity[e];
            v1[e] = psum[e] * (1.0f / (float)BATCH);
            m0 += v0[e]; m1 += v1[e];
        }
        m0 *= 1.0f / (float)NE;
        m1 *= 1.0f / (float)NE;
        float s0 = 0.f, s1 = 0.f;
        #pragma unroll
        for (int e = 0; e < NE; ++e) {
            float d0 = v0[e] - m0, d1 = v1[e] - m1;
            s0 += d0 * d0; s1 += d1 * d1;
        }
        float std0 = sqrtf(s0 / (float)(NE - 1));
        float std1 = sqrtf(s1 / (float)(NE - 1));
        float il = std0 / (m0 + 1e-8f); il *= il;
        float ll = std1 / (m1 + 1e-8f); ll *= ll;
        out[BATCH * NE + 2 * BATCH * kk] = 0.5f * il + 0.5f * ll;
    }
}

extern "C" void kernel_launch(void* const* d_in, const int* in_sizes, int n_in,
                              void* d_out, int out_size, void* d_ws, size_t ws_size,
                              hipStream_t stream) {
    (void)in_sizes; (void)n_in; (void)out_size; (void)ws_size;
    const float* x           = (const float*)d_in[0];
    const float* freq_emb    = (const float*)d_in[1];
    const float* gate_w      = (const float*)d_in[2];
    const float* freq_gate_w = (const float*)d_in[3];
    const float* complexity  = (const float*)d_in[4];
    const float* noise       = (const float*)d_in[5];
    const int*   kptr        = (const int*)d_in[6];
    float* out = (float*)d_out;

    float* pooled = (float*)d_ws;                 // 64*768 floats
    float* logits = pooled + BATCH * DIM;         // 64*16 floats

    pool_kernel<<<BATCH * DIM, 256, 0, stream>>>(x, pooled);
    logits_kernel<<<1, 128, 0, stream>>>(pooled, freq_emb, gate_w, freq_gate_w, logits);
    routing_kernel<<<1, 64, 0, stream>>>(logits, complexity, noise, kptr, out);
}